// DifferenceAwareOps_2396591751808
// MI455X (gfx1250) — compile-verified
//
#include <hip/hip_runtime.h>
#include <math.h>

#define T 8
#define B 8
#define C 128
#define H 64
#define W 64
#define HW 4096
#define EPS 1e-5f

typedef __attribute__((ext_vector_type(2))) float v2f;
typedef __attribute__((ext_vector_type(8))) float v8f;

// Workspace layout (float offsets).
// q : (b,t,c,hw)      8*8*128*4096 = 33,554,432
// k : (b,t,c,hw)      33,554,432
// v : (b,t,c,hw)      33,554,432
// p : qk partials (ks,bt,x,y) 4*64*128*128 = 4,194,304
// a : attn (bt,x,y)   64*128*128 = 1,048,576
// td: (b,x,t,hw)      33,554,432
#define Q_OFF   0u
#define K_OFF   33554432u
#define V_OFF   67108864u
#define P_OFF   100663296u
#define A_OFF   104857600u
#define TD_OFF  105906176u
// total = 139,460,608 floats = 557,842,432 bytes

#define WMMA_F32(a, b, c) \
  __builtin_amdgcn_wmma_f32_16x16x4_f32(false, (a), false, (b), (short)0, (c), false, false)

// ---------------------------------------------------------------------------
// Kernel 1: temporal diff + LayerNorm(T) + w_kv projection -> Q, K, V
// One thread per (b, c, hw). 16384 blocks x 256.
// ---------------------------------------------------------------------------
__global__ void prep_kernel(const float* __restrict__ x,
                            const float* __restrict__ w_kv,
                            float* __restrict__ ws) {
  __shared__ float s_w[2 * T * T];  // 128 floats
  if (threadIdx.x < 2 * T * T) s_w[threadIdx.x] = w_kv[threadIdx.x];
  __syncthreads();

  int g  = blockIdx.x * 256 + threadIdx.x;      // 0 .. 4,194,303
  int hw = g & (HW - 1);
  int c  = (g >> 12) & (C - 1);
  int b  = g >> 19;

  float v[T];
#pragma unroll
  for (int t = 0; t < T; ++t)
    v[t] = x[(size_t)((b * T + t) * C + c) * HW + hw];

  // shifted[t] = x[t-1]  (roll +1), diff = shifted - x
  float d[T];
  float mu = 0.f;
#pragma unroll
  for (int t = 0; t < T; ++t) { d[t] = v[(t + T - 1) & 7] - v[t]; mu += d[t]; }
  mu *= (1.f / T);
  float var = 0.f;
#pragma unroll
  for (int t = 0; t < T; ++t) { float e = d[t] - mu; var += e * e; }
  var *= (1.f / T);
  float rs = rsqrtf(var + EPS);
#pragma unroll
  for (int t = 0; t < T; ++t) d[t] = (d[t] - mu) * rs;

  float* qb = ws + Q_OFF;
  float* kb = ws + K_OFF;
  float* vb = ws + V_OFF;
#pragma unroll
  for (int t = 0; t < T; ++t)
    qb[(size_t)((b * T + t) * C + c) * HW + hw] = d[t];
#pragma unroll
  for (int f = 0; f < T; ++f) {
    float kf = 0.f, vf = 0.f;
#pragma unroll
    for (int t = 0; t < T; ++t) {
      kf = fmaf(d[t], s_w[f * T + t], kf);
      vf = fmaf(d[t], s_w[(f + T) * T + t], vf);
    }
    kb[(size_t)((b * T + f) * C + c) * HW + hw] = kf;
    vb[(size_t)((b * T + f) * C + c) * HW + hw] = vf;
  }
}

// ---------------------------------------------------------------------------
// Kernel 2: S = Q * K^T per (b,t).  M=N=128, K=4096, K split 4 ways.
// One wave per 32x32 output tile per K-chunk (2x2 WMMA micro-tile).
// 4096 waves -> 512 blocks x 256 threads.
// ---------------------------------------------------------------------------
__global__ void qk_kernel(const float* __restrict__ qbuf,
                          const float* __restrict__ kbuf,
                          float* __restrict__ part) {
  int wid  = blockIdx.x * 8 + (threadIdx.x >> 5);  // 0..4095
  int lane = threadIdx.x & 31;
  int bt   = wid >> 6;            // 0..63
  int rem  = wid & 63;
  int ks   = rem >> 4;            // K chunk 0..3
  int x0   = ((rem >> 2) & 3) * 32;
  int y0   = (rem & 3) * 32;

  const float* q = qbuf + (size_t)bt * C * HW;
  const float* k = kbuf + (size_t)bt * C * HW;

  int m    = lane & 15;
  int half = lane >> 4;
  int kb0i = ks * (HW / 4) + 2 * half;  // start col for this lane

  const float* qa0 = q + (size_t)(x0 + m) * HW + kb0i;
  const float* qa1 = qa0 + (size_t)16 * HW;
  const float* ka0 = k + (size_t)(y0 + m) * HW + kb0i;
  const float* ka1 = ka0 + (size_t)16 * HW;

  v8f acc00 = {}, acc01 = {}, acc10 = {}, acc11 = {};
#pragma unroll 4
  for (int kk = 0; kk < HW / 4; kk += 4) {
    v2f a0 = *(const v2f*)(qa0 + kk);
    v2f a1 = *(const v2f*)(qa1 + kk);
    v2f b0 = *(const v2f*)(ka0 + kk);
    v2f b1 = *(const v2f*)(ka1 + kk);
    acc00 = WMMA_F32(a0, b0, acc00);
    acc01 = WMMA_F32(a0, b1, acc01);
    acc10 = WMMA_F32(a1, b0, acc10);
    acc11 = WMMA_F32(a1, b1, acc11);
  }

  float* p = part + (size_t)(ks * 64 + bt) * C * C;
  int n0 = y0 + m;
#pragma unroll
  for (int j = 0; j < 8; ++j) {
    int r0 = x0 + j + 8 * half;
    p[(size_t)r0 * C + n0]             = acc00[j];
    p[(size_t)r0 * C + n0 + 16]        = acc01[j];
    p[(size_t)(r0 + 16) * C + n0]      = acc10[j];
    p[(size_t)(r0 + 16) * C + n0 + 16] = acc11[j];
  }
}

// ---------------------------------------------------------------------------
// Kernel 3: sum K-partials, scale by 1/64, softmax over y. Wave per row.
// 8192 rows -> 1024 blocks x 256.
// ---------------------------------------------------------------------------
__global__ void softmax_kernel(const float* __restrict__ part,
                               float* __restrict__ attn) {
  int wid  = blockIdx.x * 8 + (threadIdx.x >> 5);  // 0..8191
  int lane = threadIdx.x & 31;
  int bt   = wid >> 7;
  int xx   = wid & 127;

  float vals[4];
#pragma unroll
  for (int i = 0; i < 4; ++i) {
    int y = lane + i * 32;
    float s = 0.f;
#pragma unroll
    for (int ks = 0; ks < 4; ++ks)
      s += part[((size_t)(ks * 64 + bt) * C + xx) * C + y];
    vals[i] = s * (1.f / 64.f);  // (H*W)^-0.5
  }
  float mx = fmaxf(fmaxf(vals[0], vals[1]), fmaxf(vals[2], vals[3]));
#pragma unroll
  for (int off = 16; off >= 1; off >>= 1)
    mx = fmaxf(mx, __shfl_xor(mx, off, 32));
  float sum = 0.f;
#pragma unroll
  for (int i = 0; i < 4; ++i) { vals[i] = __expf(vals[i] - mx); sum += vals[i]; }
#pragma unroll
  for (int off = 16; off >= 1; off >>= 1)
    sum += __shfl_xor(sum, off, 32);
  float inv = 1.f / sum;
#pragma unroll
  for (int i = 0; i < 4; ++i)
    attn[((size_t)bt * C + xx) * C + lane + i * 32] = vals[i] * inv;
}

// ---------------------------------------------------------------------------
// Kernel 4: TD = A * V per (b,t).  M=128, N=4096, K=128.
// Wave per 32x32 tile (2x2 micro-tile). 32768 waves -> 4096 blocks x 256.
// Output layout (b*C + x, t, h, w) for the conv stage.
// ---------------------------------------------------------------------------
__global__ void td_kernel(const float* __restrict__ attn,
                          const float* __restrict__ vbuf,
                          float* __restrict__ td) {
  int wid  = blockIdx.x * 8 + (threadIdx.x >> 5);  // 0..32767
  int lane = threadIdx.x & 31;
  int bt   = wid >> 9;            // 0..63
  int rem  = wid & 511;
  int x0   = (rem >> 7) * 32;     // 0..96
  int hw0  = (rem & 127) * 32;    // 0..4064

  const float* A  = attn + (size_t)bt * C * C;
  const float* Vm = vbuf + (size_t)bt * C * HW;

  int m    = lane & 15;
  int half = lane >> 4;

  v8f acc00 = {}, acc01 = {}, acc10 = {}, acc11 = {};
#pragma unroll 4
  for (int kk = 0; kk < C; kk += 4) {
    int kb = kk + 2 * half;
    v2f a0 = *(const v2f*)(A + (size_t)(x0 + m) * C + kb);
    v2f a1 = *(const v2f*)(A + (size_t)(x0 + 16 + m) * C + kb);
    const float* vr = Vm + (size_t)kb * HW + hw0 + m;
    v2f b0, b1;
    b0.x = vr[0];   b0.y = vr[HW];
    b1.x = vr[16];  b1.y = vr[HW + 16];
    acc00 = WMMA_F32(a0, b0, acc00);
    acc01 = WMMA_F32(a0, b1, acc01);
    acc10 = WMMA_F32(a1, b0, acc10);
    acc11 = WMMA_F32(a1, b1, acc11);
  }

  int b = bt >> 3, t = bt & 7;
  int n0 = hw0 + m;
#pragma unroll
  for (int j = 0; j < 8; ++j) {
    int r0 = x0 + j + 8 * half;
    float* o0 = td + ((size_t)(b * C + r0) * T + t) * HW;
    float* o1 = td + ((size_t)(b * C + r0 + 16) * T + t) * HW;
    o0[n0]      = acc00[j];
    o0[n0 + 16] = acc01[j];
    o1[n0]      = acc10[j];
    o1[n0 + 16] = acc11[j];
  }
}

// ---------------------------------------------------------------------------
// Kernel 5: fused conv3x3 -> relu -> conv3x3 -> +x.  16x16 tile per block.
// 1024 images x 16 tiles = 16384 blocks x 256 threads.
// ---------------------------------------------------------------------------
__global__ void conv_kernel(const float* __restrict__ x,
                            const float* __restrict__ w1,
                            const float* __restrict__ w2,
                            const float* __restrict__ td,
                            float* __restrict__ out) {
  __shared__ float s_w1[T * T * 9];
  __shared__ float s_w2[T * T * 9];
  __shared__ float s_in[T][20][20];
  __shared__ float s_y1[T][18][18];

  int tid = threadIdx.x;
  for (int i = tid; i < T * T * 9; i += 256) { s_w1[i] = w1[i]; s_w2[i] = w2[i]; }

  int img  = blockIdx.x >> 4;       // b*C + c
  int tile = blockIdx.x & 15;
  int b = img >> 7, c = img & 127;
  int th0 = (tile >> 2) * 16, tw0 = (tile & 3) * 16;

  const float* tdi = td + (size_t)img * T * HW;

  // load 20x20 halo tile, zero-padded at image border
  for (int p = tid; p < 400; p += 256) {
    int ph = p / 20, pw = p % 20;
    int h = th0 - 2 + ph, w = tw0 - 2 + pw;
    bool inb = (h >= 0 && h < H && w >= 0 && w < W);
#pragma unroll
    for (int i = 0; i < T; ++i)
      s_in[i][ph][pw] = inb ? tdi[(size_t)i * HW + h * W + w] : 0.f;
  }
  __syncthreads();

  // conv1 + relu on 18x18 (y1 outside the image must be exactly 0)
  for (int p = tid; p < 324; p += 256) {
    int ph = p / 18, pw = p % 18;
    int h = th0 - 1 + ph, w = tw0 - 1 + pw;
    float acc[T] = {};
    if (h >= 0 && h < H && w >= 0 && w < W) {
#pragma unroll
      for (int i = 0; i < T; ++i)
#pragma unroll
        for (int ky = 0; ky < 3; ++ky)
#pragma unroll
          for (int kx = 0; kx < 3; ++kx) {
            float vv = s_in[i][ph + ky][pw + kx];
#pragma unroll
            for (int o = 0; o < T; ++o)
              acc[o] = fmaf(s_w1[((o * T + i) * 3 + ky) * 3 + kx], vv, acc[o]);
          }
    }
#pragma unroll
    for (int o = 0; o < T; ++o) s_y1[o][ph][pw] = fmaxf(acc[o], 0.f);
  }
  __syncthreads();

  // conv2 + residual on 16x16
  {
    int ph = tid >> 4, pw = tid & 15;
    int h = th0 + ph, w = tw0 + pw;
    float acc[T] = {};
#pragma unroll
    for (int i = 0; i < T; ++i)
#pragma unroll
      for (int ky = 0; ky < 3; ++ky)
#pragma unroll
        for (int kx = 0; kx < 3; ++kx) {
          float vv = s_y1[i][ph + ky][pw + kx];
#pragma unroll
          for (int o = 0; o < T; ++o)
            acc[o] = fmaf(s_w2[((o * T + i) * 3 + ky) * 3 + kx], vv, acc[o]);
        }
#pragma unroll
    for (int o = 0; o < T; ++o) {
      size_t oi = ((size_t)((b * T + o) * C + c)) * HW + h * W + w;
      out[oi] = x[oi] + acc[o];
    }
  }
}

// ---------------------------------------------------------------------------
extern "C" void kernel_launch(void* const* d_in, const int* in_sizes, int n_in,
                              void* d_out, int out_size, void* d_ws, size_t ws_size,
                              hipStream_t stream) {
  (void)in_sizes; (void)n_in; (void)out_size; (void)ws_size;
  const float* x    = (const float*)d_in[0];
  const float* w_kv = (const float*)d_in[1];
  const float* w1   = (const float*)d_in[2];
  const float* w2   = (const float*)d_in[3];
  float* out = (float*)d_out;
  float* ws  = (float*)d_ws;

  prep_kernel   <<<16384, 256, 0, stream>>>(x, w_kv, ws);
  qk_kernel     <<<  512, 256, 0, stream>>>(ws + Q_OFF, ws + K_OFF, ws + P_OFF);
  softmax_kernel<<< 1024, 256, 0, stream>>>(ws + P_OFF, ws + A_OFF);
  td_kernel     <<< 4096, 256, 0, stream>>>(ws + A_OFF, ws + V_OFF, ws + TD_OFF);
  conv_kernel   <<<16384, 256, 0, stream>>>(x, w1, w2, ws + TD_OFF, out);
}